// KUVAHeteroGNN_40759239639547
// MI455X (gfx1250) — compile-verified
//
#include <hip/hip_runtime.h>
#include <hip/hip_bf16.h>
#include <math.h>

#define NA   100000
#define NR   20000
#define EAA  1600000
#define EUA  800000
#define HIDN 128
#define NH   4
#define NF   32
#define NCLS 16

typedef __attribute__((ext_vector_type(2))) float v2f;
typedef __attribute__((ext_vector_type(8))) float v8f;

// ---------------------------------------------------------------- utilities
__global__ void fill_f32(float* __restrict__ p, float v, long n) {
  long i = (long)blockIdx.x * blockDim.x + threadIdx.x;
  long stride = (long)gridDim.x * blockDim.x;
  for (; i < n; i += stride) p[i] = v;
}

__device__ __forceinline__ float lrelu02(float x) { return x > 0.f ? x : 0.2f * x; }
__device__ __forceinline__ float gelu_exact(float x) {
  return 0.5f * x * (1.f + erff(x * 0.70710678118654752f));
}

// ---------------------------------------------------------------- WMMA GEMM
// C[nrows,128] = X[nrows,K] @ W[128,K]^T (+bias). fp32 WMMA 16x16x4.
// One wave -> 16-row x 128-col strip. Layouts per CDNA5 ISA 7.12.2:
//   A 16x4 f32: lanes 0-15 row M=lane, VGPR0=K0,VGPR1=K1; lanes 16-31 K2,K3.
//   B 4x16 f32: mirrored (N in lanes, K split across VGPRs/halves).
//   C/D 8 VGPRs: VGPR v -> M = v + 8*(lane>=16), N = lane&15.
template <int K>
__global__ void __launch_bounds__(256) gemm_xwt_wmma(
    const float* __restrict__ X, const float* __restrict__ W,
    const float* __restrict__ bias, float* __restrict__ C, int nrows) {
  const int NT = 8;  // 128 / 16 column tiles
  int wave = (int)((blockIdx.x * blockDim.x + threadIdx.x) >> 5);
  int lane = threadIdx.x & 31;
  int row0 = wave * 16;
  if (row0 >= nrows) return;
  int half = lane >> 4;
  int r = lane & 15;

  int arow = row0 + r;
  if (arow > nrows - 1) arow = nrows - 1;  // clamp; garbage rows masked at store
  const float* xrow = X + (long)arow * K;

  v8f acc[NT] = {};

  for (int k = 0; k < K; k += 4) {
    v2f a;
    a[0] = xrow[k + 2 * half + 0];
    a[1] = xrow[k + 2 * half + 1];
    const float* wbase = W + (long)r * K + k + 2 * half;
#pragma unroll
    for (int t = 0; t < NT; t++) {
      v2f b;
      b[0] = wbase[(long)t * 16 * K + 0];
      b[1] = wbase[(long)t * 16 * K + 1];
      acc[t] = __builtin_amdgcn_wmma_f32_16x16x4_f32(
          /*neg_a=*/false, a, /*neg_b=*/false, b,
          /*c_mod=*/(short)0, acc[t], /*reuse_a=*/false, /*reuse_b=*/false);
    }
  }

#pragma unroll
  for (int t = 0; t < NT; t++) {
#pragma unroll
    for (int v = 0; v < 8; v++) {
      int rr = row0 + v + 8 * half;
      if (rr < nrows) {
        int cc = t * 16 + r;
        float val = acc[t][v];
        if (bias) val += bias[cc];
        C[(long)rr * HIDN + cc] = val;
      }
    }
  }
}

// ---------------------------------------------------------------- edge passes
// pass 1: logit[e,h] = sum_f lrelu(el[src,h,f]+er[dst,h,f]) * attn[h,f]; atomic max per (dst,h)
__global__ void __launch_bounds__(256) edge_logits_k(
    const float* __restrict__ el, const float* __restrict__ er,
    const float* __restrict__ attn, const int* __restrict__ src,
    const int* __restrict__ dst, float* __restrict__ logit,
    float* __restrict__ mbuf, int nedges) {
  int idx = blockIdx.x * blockDim.x + threadIdx.x;
  if (idx >= nedges * NH) return;
  int e = idx >> 2, h = idx & 3;
  int s = src[e], d = dst[e];
  const float4* pl = (const float4*)(el + (long)s * HIDN + h * NF);
  const float4* pr = (const float4*)(er + (long)d * HIDN + h * NF);
  const float4* pa = (const float4*)(attn + h * NF);
  float acc = 0.f;
#pragma unroll
  for (int i = 0; i < NF / 4; i++) {
    float4 a = pl[i], b = pr[i], w = pa[i];
    acc += lrelu02(a.x + b.x) * w.x + lrelu02(a.y + b.y) * w.y +
           lrelu02(a.z + b.z) * w.z + lrelu02(a.w + b.w) * w.w;
  }
  logit[idx] = acc;
  int* mi = (int*)(mbuf + (long)d * NH + h);
  if (acc >= 0.f) atomicMax(mi, __float_as_int(acc));
  else            atomicMin((unsigned int*)mi, __float_as_uint(acc));
}

// pass 2: ex = exp(logit - m); atomic add per (dst,h); ex stored in-place
__global__ void __launch_bounds__(256) edge_exp_k(
    const int* __restrict__ dst, const float* __restrict__ mbuf,
    float* __restrict__ logit, float* __restrict__ sbuf, int nedges) {
  int idx = blockIdx.x * blockDim.x + threadIdx.x;
  if (idx >= nedges * NH) return;
  int e = idx >> 2, h = idx & 3;
  int d = dst[e];
  float ex = __expf(logit[idx] - mbuf[(long)d * NH + h]);
  logit[idx] = ex;
  atomicAdd(&sbuf[(long)d * NH + h], ex);
}

// pass 3: wave per edge, lane = feature within head; coalesced f32 atomic scatter
__global__ void __launch_bounds__(256) edge_scatter_k(
    const float* __restrict__ el, const float* __restrict__ exb,
    const float* __restrict__ sbuf, const int* __restrict__ src,
    const int* __restrict__ dst, float* __restrict__ out, int nedges) {
  int wave = (int)((blockIdx.x * blockDim.x + threadIdx.x) >> 5);
  int lane = threadIdx.x & 31;
  if (wave >= nedges) return;
  int s = src[wave], d = dst[wave];
#pragma unroll
  for (int h = 0; h < NH; h++) {
    float a = exb[(long)wave * NH + h] / sbuf[(long)d * NH + h];
    float v = el[(long)s * HIDN + h * NF + lane] * a;
    atomicAdd(&out[(long)d * HIDN + h * NF + lane], v);
  }
}

// ---------------------------------------------------------------- finalize
// agents: two convs each add residual h_dst and a bias -> 2*ha + biasC + biasM
__global__ void __launch_bounds__(256) finalize_agent_k(
    const float* __restrict__ acc, const float* __restrict__ hin,
    const float* __restrict__ b1, const float* __restrict__ b2,
    float* __restrict__ hout, long n) {
  long i = (long)blockIdx.x * blockDim.x + threadIdx.x;
  if (i >= n) return;
  int c = (int)(i & (HIDN - 1));
  hout[i] = gelu_exact(acc[i] + 2.f * hin[i] + b1[c] + b2[c]);
}

__global__ void __launch_bounds__(256) finalize_res_k(
    const float* __restrict__ acc, const float* __restrict__ hin,
    const float* __restrict__ b, float* __restrict__ hout, long n) {
  long i = (long)blockIdx.x * blockDim.x + threadIdx.x;
  if (i >= n) return;
  int c = (int)(i & (HIDN - 1));
  hout[i] = gelu_exact(acc[i] + hin[i] + b[c]);
}

// ---------------------------------------------------------------- readout
__global__ void pool_k(const float* __restrict__ ha, float* __restrict__ pooled, int n) {
  int c = threadIdx.x;  // 128 threads
  float local = 0.f;
  for (int row = blockIdx.x; row < n; row += gridDim.x)
    local += ha[(long)row * HIDN + c];
  atomicAdd(&pooled[c], local);
}

__global__ void __launch_bounds__(256) cls_k(
    const float* __restrict__ pooled, const float* __restrict__ w1,
    const float* __restrict__ b1, const float* __restrict__ w2,
    const float* __restrict__ b2, float* __restrict__ out, float invN) {
  __shared__ float p[HIDN];
  __shared__ float h1[2 * HIDN];
  int t = threadIdx.x;
  if (t < HIDN) p[t] = pooled[t] * invN;
  __syncthreads();
  float acc = b1[t];
  for (int k = 0; k < HIDN; k++) acc += p[k] * w1[(long)t * HIDN + k];
  h1[t] = acc > 0.f ? acc : 0.f;
  __syncthreads();
  if (t < NCLS) {
    float o = b2[t];
    for (int k = 0; k < 2 * HIDN; k++) o += h1[k] * w2[(long)t * 2 * HIDN + k];
    out[t] = o;
  }
}

// ---------------------------------------------------------------- launch
extern "C" void kernel_launch(void* const* d_in, const int* in_sizes, int n_in,
                              void* d_out, int out_size, void* d_ws, size_t ws_size,
                              hipStream_t stream) {
  (void)in_sizes; (void)n_in; (void)out_size; (void)ws_size;

  const float* agent_f    = (const float*)d_in[0];
  const float* res_f      = (const float*)d_in[1];
  const int*   collab_src = (const int*)d_in[2];
  const int*   collab_dst = (const int*)d_in[3];
  const int*   comm_src   = (const int*)d_in[4];
  const int*   comm_dst   = (const int*)d_in[5];
  const int*   uses_src   = (const int*)d_in[6];
  const int*   uses_dst   = (const int*)d_in[7];
  // params flattened as JAX tree leaves (dict keys alphabetical):
  const float* ada_b  = (const float*)d_in[8];   // adapter_agent_b
  const float* ada_w  = (const float*)d_in[9];   // adapter_agent_w [128,128]
  const float* adr_b  = (const float*)d_in[10];  // adapter_resource_b
  const float* adr_w  = (const float*)d_in[11];  // adapter_resource_w [128,64]
  const float* clsb1  = (const float*)d_in[12];
  const float* clsb2  = (const float*)d_in[13];
  const float* clsw1  = (const float*)d_in[14];  // [256,128]
  const float* clsw2  = (const float*)d_in[15];  // [16,256]
  // layers[L] relations (collaborates=0, communicates=1, uses=2),
  // leaves per relation: 0=attn, 1=bias, 2=w_dst, 3=w_src
  auto relp = [&](int L, int R, int leaf) -> const float* {
    return (const float*)d_in[16 + L * 12 + R * 4 + leaf];
  };

  float* ws = (float*)d_ws;
  long off = 0;
  auto alloc = [&](long n) { float* p = ws + off; off += n; return p; };
  float* haA   = alloc((long)NA * HIDN);
  float* haB   = alloc((long)NA * HIDN);
  float* hrA   = alloc((long)NR * HIDN);
  float* hrB   = alloc((long)NR * HIDN);
  float* el    = alloc((long)NA * HIDN);
  float* er    = alloc((long)NA * HIDN);
  float* outA  = alloc((long)NA * HIDN);
  float* outR  = alloc((long)NR * HIDN);
  float* logit = alloc((long)EAA * NH);
  float* mbuf  = alloc((long)NA * NH);
  float* sbuf  = alloc((long)NA * NH);
  float* pooled = alloc(HIDN);

  auto fill = [&](float* p, float v, long n) {
    int blocks = (int)((n + 1023) / 1024);
    if (blocks > 4096) blocks = 4096;
    if (blocks < 1) blocks = 1;
    fill_f32<<<blocks, 256, 0, stream>>>(p, v, n);
  };
  auto gemm128 = [&](const float* X, const float* W, const float* b, float* C, int nrows) {
    int waves = (nrows + 15) / 16;
    gemm_xwt_wmma<128><<<(waves + 7) / 8, 256, 0, stream>>>(X, W, b, C, nrows);
  };
  auto gemm64 = [&](const float* X, const float* W, const float* b, float* C, int nrows) {
    int waves = (nrows + 15) / 16;
    gemm_xwt_wmma<64><<<(waves + 7) / 8, 256, 0, stream>>>(X, W, b, C, nrows);
  };
  auto run_edges = [&](const float* elp, const float* erp, const float* attn,
                       const int* esrc, const int* edst, float* outacc,
                       int ndst, int nedges) {
    fill(mbuf, -INFINITY, (long)ndst * NH);
    fill(sbuf, 0.f, (long)ndst * NH);
    int tgrid = (nedges * NH + 255) / 256;
    edge_logits_k<<<tgrid, 256, 0, stream>>>(elp, erp, attn, esrc, edst, logit, mbuf, nedges);
    edge_exp_k<<<tgrid, 256, 0, stream>>>(edst, mbuf, logit, sbuf, nedges);
    edge_scatter_k<<<(nedges + 7) / 8, 256, 0, stream>>>(elp, logit, sbuf, esrc, edst, outacc, nedges);
  };

  // adapters
  gemm128(agent_f, ada_w, ada_b, haA, NA);
  gemm64(res_f, adr_w, adr_b, hrA, NR);

  float* cur_a = haA; float* nxt_a = haB;
  float* cur_r = hrA; float* nxt_r = hrB;

  for (int L = 0; L < 2; L++) {
    // agent-destination relations (collaborates + communicates), summed accumulator
    fill(outA, 0.f, (long)NA * HIDN);
    for (int R = 0; R < 2; R++) {
      const int* esrc = (R == 0) ? collab_src : comm_src;
      const int* edst = (R == 0) ? collab_dst : comm_dst;
      gemm128(cur_a, relp(L, R, 3), nullptr, el, NA);  // w_src
      gemm128(cur_a, relp(L, R, 2), nullptr, er, NA);  // w_dst
      run_edges(el, er, relp(L, R, 0), esrc, edst, outA, NA, EAA);
    }
    finalize_agent_k<<<(int)(((long)NA * HIDN + 255) / 256), 256, 0, stream>>>(
        outA, cur_a, relp(L, 0, 1), relp(L, 1, 1), nxt_a, (long)NA * HIDN);

    // uses relation (agents -> resources)
    gemm128(cur_a, relp(L, 2, 3), nullptr, el, NA);
    gemm128(cur_r, relp(L, 2, 2), nullptr, er, NR);
    fill(outR, 0.f, (long)NR * HIDN);
    run_edges(el, er, relp(L, 2, 0), uses_src, uses_dst, outR, NR, EUA);
    finalize_res_k<<<(int)(((long)NR * HIDN + 255) / 256), 256, 0, stream>>>(
        outR, cur_r, relp(L, 2, 1), nxt_r, (long)NR * HIDN);

    float* t;
    t = cur_a; cur_a = nxt_a; nxt_a = t;
    t = cur_r; cur_r = nxt_r; nxt_r = t;
  }

  // readout + classifier head
  fill(pooled, 0.f, HIDN);
  pool_k<<<512, 128, 0, stream>>>(cur_a, pooled, NA);
  cls_k<<<1, 256, 0, stream>>>(pooled, clsw1, clsb1, clsw2, clsb2,
                               (float*)d_out, 1.f / (float)NA);
}